// Homography_Loss_69647189672374
// MI455X (gfx1250) — compile-verified
//
#include <hip/hip_runtime.h>
#include <math.h>

typedef __attribute__((ext_vector_type(2))) float v2f;
typedef __attribute__((ext_vector_type(8))) float v8f;

#define NB   64
#define NM   64
#define NPTS 320   // points per batch (M*5)

__constant__ float c_cls_mean[9] = {
    1.76255119f,     0.66068622f,     0.84422524f,
    1.52563191462f,  1.62856739989f,  3.88311640418f,
    1.73698127f,     0.59706367f,     1.76282397f };

__device__ __forceinline__ float smooth_l1(float d) {
    float ad = fabsf(d);
    return ad < 1.0f ? 0.5f * d * d : ad - 0.5f;
}

__global__ __launch_bounds__(32)
void homography_loss_kernel(
    const float* __restrict__ centers,    // (K,2)
    const int*   __restrict__ cls_ids,    // (K)
    const float* __restrict__ ry,         // (K)
    const float* __restrict__ size_3d,    // (K,3)
    const float* __restrict__ Pgt,        // (K,5,2)
    const float* __restrict__ Qgt,        // (K,5,3)
    const float* __restrict__ off_gt,     // (K,2)
    const float* __restrict__ depth_gt,   // (K)
    const float* __restrict__ out_off,    // (K,2)
    const float* __restrict__ out_depth,  // (K)
    const float* __restrict__ calibs,     // (B,3,4)
    const float* __restrict__ trans_inv,  // (B,2,3)
    float* __restrict__ ws)               // 3 combos * 4 accumulators
{
    __shared__ float su[NPTS], sv[NPTS];   // dst (Qpred x, Qpred z)
    __shared__ float sx[NPTS], sy[NPTS];   // src (Pgt x, y)
    __shared__ float g[9 * 16];            // G = A^T A (9x9, padded rows)
    __shared__ float xv[16], yv[16], rs[16];

    const int lane  = threadIdx.x;
    const int b     = blockIdx.x & 63;
    const int combo = blockIdx.x >> 6;

    // ----- phase 1: per-box dst points -----
    const float b1c[4] = { 0.5f, -0.5f, -0.5f,  0.5f };  // BASE_CORNERS row1 (first 4)
    const float b2c[4] = { 0.5f,  0.5f, -0.5f, -0.5f };  // BASE_CORNERS row2 (first 4)
    const float* cb = calibs + b * 12;
    const float f0 = cb[0], c0 = cb[2];
    const float t0 = cb[3] / (-f0);
    const float* ti = trans_inv + b * 6;

    for (int r = 0; r < 2; ++r) {
        int m = lane + r * 32;
        int k = b * NM + m;
        int cls = cls_ids[k];
        float d1 = size_3d[k * 3 + 1] + c_cls_mean[cls * 3 + 1];
        float d2 = size_3d[k * 3 + 2] + c_cls_mean[cls * 3 + 2];
        float cs = cosf(ry[k]);
        float sn = sinf(ry[k]);
        float offx, offy, dep;
        if (combo == 0)      { offx = off_gt[k*2];  offy = off_gt[k*2+1];  dep = out_depth[k]; }
        else if (combo == 1) { offx = out_off[k*2]; offy = out_off[k*2+1]; dep = depth_gt[k]; }
        else                 { offx = out_off[k*2]; offy = out_off[k*2+1]; dep = out_depth[k]; }
        float ux = (centers[k*2]   + offx) * 4.0f;
        float uy = (centers[k*2+1] + offy) * 4.0f;
        float pjx = ti[0]*ux + ti[1]*uy + ti[2];
        float p3x = (pjx - c0) * dep / f0 + t0;   // Qpred x of center point
        float p3z = dep;                          // Qpred z of center point
        su[m*5 + 0] = p3x;
        sv[m*5 + 0] = p3z;
        #pragma unroll
        for (int j = 0; j < 4; ++j) {
            // scaled row0 = BASE[2]*d2, row2 = BASE[1]*d1; corner_x/z via R(ry)
            float s0 = b2c[j] * d2;
            float s2 = b1c[j] * d1;
            su[m*5 + 1 + j] = p3x + ( cs * s0 + sn * s2);
            sv[m*5 + 1 + j] = p3z + (-sn * s0 + cs * s2);
        }
    }
    // src points
    for (int i = 0; i < 10; ++i) {
        int n = lane + 32 * i;
        sx[n] = Pgt[(b * NPTS + n) * 2 + 0];
        sy[n] = Pgt[(b * NPTS + n) * 2 + 1];
    }
    __syncthreads();

    // ----- phase 2: G = A^T A via 160 x v_wmma_f32_16x16x4_f32 -----
    // f32 16x4 A-operand layout: lane L, vgpr p -> (M = L&15, K = p + 2*(L>=16)).
    // Both operands of A^T*A want identical per-lane values -> one fragment
    // feeds SRC0 & SRC1.  Column selectors are loop-invariant: hoist them as
    // 0/±1 coefficients so the inner loop is branchless FMAs (EXEC stays ~0
    // -mask-churn-free around every WMMA).
    const int hi  = lane >> 4;
    const int col = lane & 15;
    const float pe_x = (col == 0) ? 1.0f : 0.0f;   // even row: x
    const float pe_y = (col == 1) ? 1.0f : 0.0f;   // even row: y
    const float pe_1 = (col == 2) ? 1.0f : 0.0f;   // even row: 1
    const float po_x = (col == 3) ? 1.0f : 0.0f;   // odd row: x
    const float po_y = (col == 4) ? 1.0f : 0.0f;   // odd row: y
    const float po_1 = (col == 5) ? 1.0f : 0.0f;   // odd row: 1
    const float pm_x = (col == 6) ? -1.0f : 0.0f;  // cols 6..8: -(x,y,1)*{u|v}
    const float pm_y = (col == 7) ? -1.0f : 0.0f;
    const float pm_1 = (col == 8) ? -1.0f : 0.0f;

    v8f acc = {0.f, 0.f, 0.f, 0.f, 0.f, 0.f, 0.f, 0.f};
    #pragma unroll 4
    for (int t = 0; t < 160; ++t) {
        int n = 2 * t + hi;
        float x = sx[n], y = sy[n], u = su[n], v = sv[n];
        float w  = fmaf(pm_x, x, fmaf(pm_y, y, pm_1));   // shared -(x|y|1) term
        float be = fmaf(pe_x, x, fmaf(pe_y, y, pe_1));
        float bo = fmaf(po_x, x, fmaf(po_y, y, po_1));
        v2f frag;
        frag.x = fmaf(u, w, be);   // vgpr0: even row of point n (row 4t+2*hi)
        frag.y = fmaf(v, w, bo);   // vgpr1: odd  row of point n (row 4t+2*hi+1)
        acc = __builtin_amdgcn_wmma_f32_16x16x4_f32(
            false, frag, false, frag, (short)0, acc, false, false);
    }
    // D layout: vgpr p, lanes 0-15 -> M=p; lanes 16-31 -> M=p+8; N = lane&15
    #pragma unroll
    for (int p = 0; p < 8; ++p) {
        int Mr = p + hi * 8;
        if (Mr < 9 && col < 9) g[Mr * 16 + col] = acc[p];
    }
    __syncthreads();

    // ----- phase 3: min-eigenvector of G via shifted power iteration -----
    if (lane < 9) {
        float rsum = g[lane * 16 + lane];
        for (int j = 0; j < 9; ++j)
            if (j != lane) rsum += fabsf(g[lane * 16 + j]);
        rs[lane] = rsum;
    }
    if (lane < 16) xv[lane] = 1.0f;
    __syncthreads();
    float s = rs[0];
    for (int j = 1; j < 9; ++j) s = fmaxf(s, rs[j]);   // Gershgorin upper bound

    for (int it = 0; it < 96; ++it) {
        float yval = 0.0f;
        if (lane < 9) {
            yval = s * xv[lane];
            for (int j = 0; j < 9; ++j) yval -= g[lane * 16 + j] * xv[j];
        }
        __syncthreads();
        if (lane < 9) yv[lane] = yval;
        __syncthreads();
        float nrm = 1e-30f;
        for (int j = 0; j < 9; ++j) nrm += yv[j] * yv[j];
        float inv = rsqrtf(nrm);
        __syncthreads();
        if (lane < 9) xv[lane] = yv[lane] * inv;
        __syncthreads();
    }
    // h = xv[0..8];  H[j][i] = h[3j+i]  (sign/scale cancels in the ratio)

    // ----- phase 4: reprojection + masked smooth-L1 -----
    float hn = 0.f, hd = 0.f, rn = 0.f, rd = 0.f;
    for (int i = 0; i < 10; ++i) {
        int n = lane + 32 * i;
        float x = sx[n], y = sy[n];
        float n0 = xv[0] * x + xv[1] * y + xv[2];
        float n1 = xv[3] * x + xv[4] * y + xv[5];
        float n2 = xv[6] * x + xv[7] * y + xv[8];
        float iz = 1.0f / (n2 + 1e-10f);
        float r0 = n0 * iz, r1 = n1 * iz;
        bool msk = (r0 > -40.f) && (r0 < 40.f) && (r1 > 0.f) && (r1 < 80.f);
        float q0 = Qgt[(b * NPTS + n) * 3 + 0];
        float q2 = Qgt[(b * NPTS + n) * 3 + 2];
        if (msk) {
            hn += smooth_l1(r0 - q0) + smooth_l1(r1 - q2);
            hd += 2.0f;
        } else {
            rn += smooth_l1(su[n] - q0) + smooth_l1(sv[n] - q2);
            rd += 2.0f;
        }
    }
    #pragma unroll
    for (int off = 16; off > 0; off >>= 1) {
        hn += __shfl_down(hn, off, 32);
        hd += __shfl_down(hd, off, 32);
        rn += __shfl_down(rn, off, 32);
        rd += __shfl_down(rd, off, 32);
    }
    if (lane == 0) {
        atomicAdd(&ws[combo * 4 + 0], hn);
        atomicAdd(&ws[combo * 4 + 1], hd);
        atomicAdd(&ws[combo * 4 + 2], rn);
        atomicAdd(&ws[combo * 4 + 3], rd);
    }
}

__global__ void homography_finalize_kernel(const float* __restrict__ ws,
                                           float* __restrict__ out)
{
    if (threadIdx.x == 0 && blockIdx.x == 0) {
        float tot = 0.0f;
        for (int c = 0; c < 3; ++c) {
            float homo   = ws[c * 4 + 0] / fmaxf(ws[c * 4 + 1], 1.0f);
            float reproj = ws[c * 4 + 2] / fmaxf(ws[c * 4 + 3], 1.0f);
            tot += homo + 0.5f * reproj;
        }
        out[0] = tot / 3.0f;
    }
}

extern "C" void kernel_launch(void* const* d_in, const int* in_sizes, int n_in,
                              void* d_out, int out_size, void* d_ws, size_t ws_size,
                              hipStream_t stream) {
    // dict order: mask_2d, center_3dto2d, cls_ids, ry, size_3d, Pgt, Qgt,
    //             offset_3d_gt, depth_gt, out_offset_3d, out_depth, calibs, trans_inv
    const float* centers   = (const float*)d_in[1];
    const int*   cls_ids   = (const int*)  d_in[2];
    const float* ryp       = (const float*)d_in[3];
    const float* size_3d   = (const float*)d_in[4];
    const float* Pgt       = (const float*)d_in[5];
    const float* Qgt       = (const float*)d_in[6];
    const float* off_gt    = (const float*)d_in[7];
    const float* depth_gt  = (const float*)d_in[8];
    const float* out_off   = (const float*)d_in[9];
    const float* out_depth = (const float*)d_in[10];
    const float* calibs    = (const float*)d_in[11];
    const float* trans_inv = (const float*)d_in[12];
    float* ws  = (float*)d_ws;
    float* out = (float*)d_out;

    hipMemsetAsync(ws, 0, 12 * sizeof(float), stream);
    homography_loss_kernel<<<dim3(3 * NB), dim3(32), 0, stream>>>(
        centers, cls_ids, ryp, size_3d, Pgt, Qgt, off_gt, depth_gt,
        out_off, out_depth, calibs, trans_inv, ws);
    homography_finalize_kernel<<<dim3(1), dim3(32), 0, stream>>>(ws, out);
}